// CameraLiDARMutualInformation_57569741635836
// MI455X (gfx1250) — compile-verified
//
#include <hip/hip_runtime.h>
#include <hip/hip_bf16.h>
#include <math.h>

// CDNA5 / gfx1250 camera-LiDAR mutual-information projection kernel.
// 16 points per wave32; 81-tap Gaussian gather reduced via a chain of
// V_WMMA_F32_16X16X4_F32 (diagonal of V * W^T), denominator via shfl_xor.

#define NPTS 300000
#define IMH  1080
#define IMW  1920

typedef __attribute__((ext_vector_type(2))) float v2f;
typedef __attribute__((ext_vector_type(8))) float v8f;

__global__ __launch_bounds__(256) void cam_lidar_mi_kernel(
    const float* __restrict__ rPLl,   // (3, N) row-major: [0..N), [N..2N), [2N..3N)
    const float* __restrict__ inten,  // (1, N)
    const float* __restrict__ image,  // (H, W)
    const float* __restrict__ eta,    // (6, 1)
    const float* __restrict__ Kc,     // (3, 3)
    const float* __restrict__ dist,   // (5,)
    float* __restrict__ out)          // [N] I_lidar, [N] I_image
{
    const int lane = threadIdx.x & 31;
    const int wid  = threadIdx.x >> 5;
    const int gw   = blockIdx.x * (blockDim.x >> 5) + wid;
    const int base = gw * 16;
    if (base >= NPTS) return;          // wave-uniform: EXEC stays all-ones below

    const int m    = lane & 15;        // point-in-group (both A's M and B's N index)
    const int half = lane >> 4;        // selects K sub-pair per f32 WMMA layout
    const int p    = base + m;         // N divisible by 16 -> always in range here

    // ---- uniform constants (scalarized loads) ----
    const float t0c = eta[0], t1c = eta[1], t2c = eta[2];
    const float s0  = eta[3], s1  = eta[4], s2  = eta[5];
    const float fx  = Kc[0], cx = Kc[2], fy = Kc[4], cy = Kc[5];
    const float k1  = dist[0], k2 = dist[1], pp1 = dist[2], pp2 = dist[3], k3 = dist[4];

    // ---- Rodrigues: R = I + sin(th)*Kx + (1-cos(th))*Kx^2 ----
    const float th  = sqrtf(s0*s0 + s1*s1 + s2*s2 + 1e-24f);
    const float rki = 1.0f / th;
    const float kx = s0*rki, ky = s1*rki, kz = s2*rki;
    const float sn = __sinf(th), c1 = 1.0f - __cosf(th);
    const float K00 = -(kz*kz + ky*ky), K01 = kx*ky, K02 = kx*kz;
    const float K11 = -(kz*kz + kx*kx), K12 = ky*kz, K22 = -(ky*ky + kx*kx);
    const float R00 = 1.0f + c1*K00, R01 = -sn*kz + c1*K01, R02 =  sn*ky + c1*K02;
    const float R10 =  sn*kz + c1*K01, R11 = 1.0f + c1*K11, R12 = -sn*kx + c1*K12;
    const float R20 = -sn*ky + c1*K02, R21 =  sn*kx + c1*K12, R22 = 1.0f + c1*K22;

    // ---- per-point projection ----
    const float px = rPLl[p], py = rPLl[NPTS + p], pz = rPLl[2*NPTS + p];
    const float inten_p = inten[p];
    const bool considered = sqrtf(px*px + py*py + pz*pz) < 15.0f;
    const float dxl = px - t0c, dyl = py - t1c, dzl = pz - t2c;
    const float X = R00*dxl + R10*dyl + R20*dzl;           // R^T * (p - t)
    const float Y = R01*dxl + R11*dyl + R21*dzl;
    const float Z = R02*dxl + R12*dyl + R22*dzl;
    const bool in_front = Z > 1e-6f;
    const float zs = in_front ? Z : 1.0f;
    const float u = X / zs, v = Y / zs;
    const float r2 = u*u + v*v;
    const float radial = 1.0f + k1*r2 + k2*r2*r2 + k3*r2*r2*r2;
    const float ud = u*radial + 2.0f*pp1*u*v + pp2*(r2 + 2.0f*u*u);
    const float vd = v*radial + pp1*(r2 + 2.0f*v*v) + 2.0f*pp2*u*v;
    const float pu = fx*ud + cx;
    const float pv = fy*vd + cy;
    const bool valid = considered && in_front &&
                       (pu >= 0.0f) && (pu <= (float)IMW) &&
                       (pv >= 0.0f) && (pv <= (float)IMH);
    const int iu0 = (int)floorf(pu);
    const int iv0 = (int)floorf(pv);

    // ---- prefetch the 9 image rows touched by this point into cache ----
    if (half == 0) {
        int ixc = iu0 - 4; ixc = ixc < 0 ? 0 : (ixc > IMW-1 ? IMW-1 : ixc);
        #pragma unroll
        for (int oy = -4; oy <= 4; ++oy) {
            int iyr = iv0 + oy; iyr = iyr < 0 ? 0 : (iyr > IMH-1 ? IMH-1 : iyr);
            __builtin_prefetch(&image[iyr * IMW + ixc], 0, 1);   // global_prefetch_b8
        }
    }

    // ---- 84-tap (81 + 3 zero-pad) accumulation via WMMA f32 16x16x4 chain ----
    // A[m, k] = image value for point m, tap k ; B[n, k] = weight for point n, tap k
    // f32 WMMA layout => this lane supplies (m = lane&15, k = 4c + 2*half + {0,1})
    // for BOTH A and B. diag(D) = per-point weighted sums.
    v8f acc = (v8f){0.f,0.f,0.f,0.f,0.f,0.f,0.f,0.f};
    float wsum = 0.0f;
    #pragma unroll 3
    for (int c = 0; c < 21; ++c) {
        const int tb = 4*c + 2*half;
        float va0 = 0.f, va1 = 0.f, wa0 = 0.f, wa1 = 0.f;
        #pragma unroll
        for (int j = 0; j < 2; ++j) {
            const int t = tb + j;
            float wv = 0.0f, vv = 0.0f;
            if (t < 81) {                       // taps 81..83 are zero padding
                const int jy = t / 9;
                const int jx = t - 9*jy;        // ox varies fastest in reference
                int ix = iu0 + (jx - 4);
                int iy = iv0 + (jy - 4);
                ix = ix < 0 ? 0 : (ix > IMW-1 ? IMW-1 : ix);
                iy = iy < 0 ? 0 : (iy > IMH-1 ? IMH-1 : iy);
                const float ddx = (float)ix - pu;
                const float ddy = (float)iy - pv;
                wv = __expf(-(ddx*ddx + ddy*ddy) * 0.125f);   // 1/(2*sigma^2), sigma=2
                vv = image[iy * IMW + ix];
            }
            if (j == 0) { va0 = vv; wa0 = wv; } else { va1 = vv; wa1 = wv; }
        }
        v2f A; A.x = va0; A.y = va1;
        v2f B; B.x = wa0; B.y = wa1;
        acc = __builtin_amdgcn_wmma_f32_16x16x4_f32(
                  false, A, false, B, (short)0, acc, false, false);
        wsum += wa0 + wa1;
    }

    // ---- denominator: sum weights across the two K-halves of each point ----
    const float wtot = wsum + __shfl_xor(wsum, 16, 32);

    // ---- extract diag(D): m<8 -> VGPR m @ lane m ; m>=8 -> VGPR m-8 @ lane m+16 ----
    const int sel = (lane < 8) ? lane : (lane - 24);
    float dsel = acc[0];
    if (sel == 1) dsel = acc[1];
    if (sel == 2) dsel = acc[2];
    if (sel == 3) dsel = acc[3];
    if (sel == 4) dsel = acc[4];
    if (sel == 5) dsel = acc[5];
    if (sel == 6) dsel = acc[6];
    if (sel == 7) dsel = acc[7];

    if (lane < 8 || lane >= 24) {      // writer lanes hold their own point's data (pt == p)
        const float mk = valid ? 1.0f : 0.0f;
        out[p]        = inten_p * mk;              // I_lidar
        out[NPTS + p] = (dsel / wtot) * mk;        // I_image
    }
}

extern "C" void kernel_launch(void* const* d_in, const int* in_sizes, int n_in,
                              void* d_out, int out_size, void* d_ws, size_t ws_size,
                              hipStream_t stream) {
    const float* rPLl  = (const float*)d_in[0];
    const float* inten = (const float*)d_in[1];
    const float* image = (const float*)d_in[2];
    const float* eta   = (const float*)d_in[3];
    const float* Kc    = (const float*)d_in[4];
    const float* dist  = (const float*)d_in[5];
    float* out = (float*)d_out;

    const int waves  = (NPTS + 15) / 16;         // 18750
    const int blocks = (waves + 7) / 8;          // 8 waves (256 threads) per block
    cam_lidar_mi_kernel<<<blocks, 256, 0, stream>>>(rPLl, inten, image, eta, Kc, dist, out);
}